// DPPLoss_2396591751988
// MI455X (gfx1250) — compile-verified
//
#include <hip/hip_runtime.h>
#include <hip/hip_bf16.h>
#include <math.h>

// ---------------------------------------------------------------------------
// DPP loss via Weinstein–Aronszajn: det(I_N + K K^T) = det(I_D + K^T K).
// Heavy lifting = two D x D Gram matrices computed with V_WMMA_F32_16X16X4_F32.
// ---------------------------------------------------------------------------

typedef __attribute__((ext_vector_type(2))) float v2f;
typedef __attribute__((ext_vector_type(8))) float v8f;

#define DPP_D 256
#define DPP_EPS 1e-5f
#define DPP_LOG_EPS (-11.512925464970229)   // log(1e-5) in double

// ---------------------------------------------------------------------------
// Kernel 1: selection mask m_i = (y_i > 0.5) ? 1 : 0
// ---------------------------------------------------------------------------
__global__ void dpp_mask_kernel(const float* __restrict__ y,
                                float* __restrict__ mask, int n) {
    int i = blockIdx.x * blockDim.x + threadIdx.x;
    if (i < n) mask[i] = (y[i] > 0.5f) ? 1.0f : 0.0f;
}

// ---------------------------------------------------------------------------
// Kernel 2: G_full = K^T K  and  G_sel = K^T diag(m) K, both D x D (D=256),
// tiled 16x16, one wave (32 threads) per tile, fp32 WMMA along N.
//
// A-matrix 16x4 f32 layout (ISA 7.12.2): lane L holds row M = L%16;
//   VGPR0 = K-col {0 | 2} for lanes {0-15 | 16-31}, VGPR1 = K-col {1 | 3}.
// B-matrix 4x16 layout mirrored (row kk striped across lanes within a VGPR).
// C/D 16x16 f32: VGPR v -> row v + 8*(lane>=16), col = lane%16.
// ---------------------------------------------------------------------------
__global__ void dpp_gram_wmma(const float* __restrict__ K,   // [N][256] row-major
                              const float* __restrict__ mask,// [N]
                              float* __restrict__ Gfull,     // [256][256]
                              float* __restrict__ Gsel,      // [256][256]
                              int N) {
    const int tc   = blockIdx.x;        // tile col (0..15)
    const int tr   = blockIdx.y;        // tile row (0..15)
    const int lane = threadIdx.x;       // 0..31, EXEC all ones (WMMA requirement)
    const int half = lane >> 4;         // 0 | 1
    const int l16  = lane & 15;

    const int ca = tr * 16 + l16;       // column of K feeding the A operand
    const int cb = tc * 16 + l16;       // column of K feeding the B operand

    v8f accF = {};                      // C starts at 0
    v8f accS = {};

    #pragma unroll 4
    for (int n = 0; n < N; n += 4) {
        const int r0 = n + 2 * half;    // this lane-half's two K rows
        const float* __restrict__ row0 = K + (size_t)r0 * DPP_D;
        const float* __restrict__ row1 = row0 + DPP_D;

        v2f a, b, am;
        a.x = row0[ca];  a.y = row1[ca];
        b.x = row0[cb];  b.y = row1[cb];

        const float m0 = mask[r0];
        const float m1 = mask[r0 + 1];
        am.x = a.x * m0; am.y = a.y * m1;   // mask on A only (m^2 == m)

        // D = A * B + C   (v_wmma_f32_16x16x4_f32)
        accF = __builtin_amdgcn_wmma_f32_16x16x4_f32(
            false, a,  false, b, (short)0, accF, false, false);
        accS = __builtin_amdgcn_wmma_f32_16x16x4_f32(
            false, am, false, b, (short)0, accS, false, false);
    }

    #pragma unroll
    for (int v = 0; v < 8; ++v) {
        const int gr = tr * 16 + v + 8 * half;
        const int gc = tc * 16 + l16;
        Gfull[gr * DPP_D + gc] = accF[v];
        Gsel [gr * DPP_D + gc] = accS[v];
    }
}

// ---------------------------------------------------------------------------
// Kernel 3: one 256-thread block. Count |Y|, add diagonals, two in-place
// Choleskys (right-looking, lower triangle) on the 256x256 SPD matrices in
// workspace, log-det from diag, final combine in double.
// ---------------------------------------------------------------------------
__device__ float dpp_block_cholesky_logdet(float* __restrict__ A, float* red) {
    const int n = DPP_D;
    const int t = threadIdx.x;          // blockDim.x == n == 256
    for (int j = 0; j < n; ++j) {
        if (t == 0) A[j * n + j] = sqrtf(fmaxf(A[j * n + j], 1e-30f));
        __syncthreads();
        const float dj = A[j * n + j];
        if (t > j) A[t * n + j] /= dj;              // scale column j
        __syncthreads();
        if (t > j) {                                // rank-1 trailing update, row t
            const float ltj = A[t * n + j];
            for (int k = j + 1; k <= t; ++k)
                A[t * n + k] = fmaf(-ltj, A[k * n + j], A[t * n + k]);
        }
        __syncthreads();
    }
    red[t] = 2.0f * logf(A[t * n + t]);
    __syncthreads();
    for (int s = 128; s > 0; s >>= 1) {
        if (t < s) red[t] += red[t + s];
        __syncthreads();
    }
    const float r = red[0];
    __syncthreads();
    return r;
}

__global__ void dpp_logdet_loss_kernel(float* __restrict__ Gfull,
                                       float* __restrict__ Gsel,
                                       const float* __restrict__ mask,
                                       float* __restrict__ out, int N) {
    __shared__ float red[DPP_D];
    const int t = threadIdx.x;

    // |Y| = sum(mask)
    float c = 0.0f;
    for (int i = t; i < N; i += DPP_D) c += mask[i];
    red[t] = c;
    __syncthreads();
    for (int s = 128; s > 0; s >>= 1) {
        if (t < s) red[t] += red[t + s];
        __syncthreads();
    }
    const float nsel = red[0];
    __syncthreads();

    // M1 = I + K^T K ; M2 = eps*I + K_Y^T K_Y
    Gfull[t * DPP_D + t] += 1.0f;
    Gsel [t * DPP_D + t] += DPP_EPS;
    __syncthreads();

    const float ld1 = dpp_block_cholesky_logdet(Gfull, red);
    const float ld2 = dpp_block_cholesky_logdet(Gsel,  red);

    if (t == 0) {
        // det_L_I = det(I + K^T K); det >= 1 so exp() is safe at this scale.
        const double det1 = exp((double)ld1) + (double)DPP_EPS;
        // det_L_Y = eps^{|Y|-D} * det(eps I + G_sel); underflows to 0 for
        // |Y| >> D exactly like the reference's fp32 LU, leaving log(eps).
        const double lY   = ((double)nsel - (double)DPP_D) * DPP_LOG_EPS + (double)ld2;
        const double det2 = exp(lY) + (double)DPP_EPS;
        const double loss = log(det1) - log(det2);
        out[0] = (nsel > 0.0f) ? (float)loss : 0.0f;
    }
}

// ---------------------------------------------------------------------------
// Launch
// ---------------------------------------------------------------------------
extern "C" void kernel_launch(void* const* d_in, const int* in_sizes, int n_in,
                              void* d_out, int out_size, void* d_ws, size_t ws_size,
                              hipStream_t stream) {
    const float* K = (const float*)d_in[0];   // pred_k [N,256]
    const float* y = (const float*)d_in[1];   // y_true [N,1]
    float* out = (float*)d_out;

    const int N = in_sizes[1];                // 4096
    // workspace layout: mask[N] | Gfull[256*256] | Gsel[256*256]
    char*  ws    = (char*)d_ws;
    float* mask  = (float*)ws;
    float* Gfull = (float*)(ws + (size_t)N * sizeof(float));
    float* Gsel  = Gfull + DPP_D * DPP_D;

    dpp_mask_kernel<<<(N + 255) / 256, 256, 0, stream>>>(y, mask, N);

    dim3 grid(DPP_D / 16, DPP_D / 16);        // 16 x 16 tiles
    dpp_gram_wmma<<<grid, 32, 0, stream>>>(K, mask, Gfull, Gsel, N);

    dpp_logdet_loss_kernel<<<1, DPP_D, 0, stream>>>(Gfull, Gsel, mask, out, N);
}